// MonarchMatrix_78451872628910
// MI455X (gfx1250) — compile-verified
//
#include <hip/hip_runtime.h>
#include <hip/hip_bf16.h>
#include <stdint.h>

typedef float v2f __attribute__((ext_vector_type(2)));
typedef float v8f __attribute__((ext_vector_type(8)));
typedef unsigned int v4u __attribute__((ext_vector_type(4)));
typedef int v8i __attribute__((ext_vector_type(8)));
typedef int v4i __attribute__((ext_vector_type(4)));

// Monarch: out = P(R_blk(P(L_blk(P x)))) + bias, S=128, n=16384.
// Phase 1: W1[t][n][b] = sum_m L[b][n][m] * x[t][m*128+b]      (A strided, b-grouped)
// Phase 2: out[t][p][q] = sum_m R[q][p][m] * W1[t][q*128+m] + bias[p*128+q]
//
// Workgroup tile: 16 tokens x 16 n x 32 blocks; 8 waves, each wave owns 4 blocks
// with one 16x16 f32 WMMA accumulator per block. K chunked by 8 through LDS.
// x/W1 tiles:   gfx1250 async global->LDS loads (ASYNCcnt, deep MLP for HBM streams)
// weight tiles: gfx1250 Tensor Data Mover, one 3D-tile descriptor per chunk
//               (TENSORcnt; TDM LDS padding reproduces the fragment layout).

#define SD 128
#define NF 16384
#define ROWS 12    // floats per LDS row (8 data + 4 pad) -> 48B rows, 16B aligned
#define ABSTR 192  // per-block stride in A/W chunk buffers: 16 rows * 12 floats
#define CBSTR 273  // per-block stride for C staging (odd -> conflict-free transpose)

template <int PHASE>
__global__ __launch_bounds__(256) void monarch_phase(const float* __restrict__ A,
                                                     const float* __restrict__ Wgt,
                                                     const float* __restrict__ bias,
                                                     float* __restrict__ Out) {
  // Two chunk buffers (A, W), 32 blocks * 192 floats each = 24KB each.
  // C staging (32*273 = 8736 floats) reuses the same LDS after the K loop.
  __shared__ float lds[2 * 32 * ABSTR];  // 12288 floats = 49152 B

  const int tid = threadIdx.x;
  const int wave = tid >> 5;
  const int lane = tid & 31;
  const int laneM = lane & 15;        // row (M for A, N for B, N for C)
  const int laneK = (lane >> 4) * 2;  // k base offset within a 4-wide k-step

  const int t0 = blockIdx.x * 16;  // token tile
  const int n0 = blockIdx.y * 16;  // output-within-block tile
  const int b0 = blockIdx.z * 32;  // block group (b for phase1, q for phase2)

  // LDS byte addresses for async-to-LDS / TDM destination operands
  const unsigned alds_base = (unsigned)(uintptr_t)(&lds[0]);
  const unsigned wlds_base = alds_base + 32u * ABSTR * 4u;

  v8f acc[4] = {};  // 4 blocks per wave, 16x16 f32 accumulator each

  for (int mc = 0; mc < 16; ++mc) {  // K loop, chunks of 8
    const int m0 = mc * 8;
    __syncthreads();  // previous chunk's LDS reads complete before overwrite

    // ---- async stage A chunk -> lds[bl*ABSTR + tt*ROWS + mm]
    if (PHASE == 1) {
      // x[t][m][b]: b fastest -> per-lane B32 gather, 32 lanes = one 128B line
      const int bl = tid & 31, r = tid >> 5;
#pragma unroll
      for (int it = 0; it < 16; ++it) {
        const int p = it * 8 + r;  // 128 (tt,mm) pairs
        const int tt = p >> 3, mm = p & 7;
        const float* gp = &A[(size_t)(t0 + tt) * NF + (size_t)(m0 + mm) * SD + (b0 + bl)];
        const unsigned la = alds_base + (unsigned)((bl * ABSTR + tt * ROWS + mm) * 4);
        asm volatile("global_load_async_to_lds_b32 %0, %1, off"
                     :: "v"(la), "v"(gp) : "memory");
      }
    } else {
      // W1[t][q][m]: m fastest -> B128 (4 floats / lane), 16B aligned
      const int mm = (tid & 1) * 4;
#pragma unroll
      for (int it = 0; it < 4; ++it) {
        const int idx = (tid >> 1) + it * 128;  // 512 (bl,tt) pairs
        const int bl = idx >> 4, tt = idx & 15;
        const float* gp = &A[(size_t)(t0 + tt) * NF + (size_t)(b0 + bl) * SD + (m0 + mm)];
        const unsigned la = alds_base + (unsigned)((bl * ABSTR + tt * ROWS + mm) * 4);
        asm volatile("global_load_async_to_lds_b128 %0, %1, off"
                     :: "v"(la), "v"(gp) : "memory");
      }
    }

    // ---- weight chunk via Tensor Data Mover (one descriptor, issued by wave 0):
    // 3D tile: x = 8 m (contiguous), y = 16 n (stride 128), z = 32 blocks (stride 16384)
    // LDS pad: 4 DWORDs after every 8 DWORDs -> row stride 12, block stride 192 = ABSTR
    if (wave == 0) {
      const uint64_t ga =
          (uint64_t)(uintptr_t)&Wgt[(size_t)b0 * NF + (size_t)n0 * SD + m0];
      v4u g0;
      g0.x = 1u;                                    // count=1, user descriptor
      g0.y = wlds_base;                             // lds_addr
      g0.z = (unsigned)ga;                          // global_addr[31:0]
      g0.w = (unsigned)(ga >> 32) | 0x80000000u;    // global_addr[56:32] | type=2
      v8i g1;
      g1[0] = 0x06920000;                 // data_size=4B, pad_en, intvl=8DW, amt=4DW
      g1[1] = (int)(128u << 16);          // tensor_dim0 = 128   (bits 79:48)
      g1[2] = (int)(16384u << 16);        // tensor_dim1 = 16384 (bits 111:80)
      g1[3] = (int)(8u << 16);            // tile_dim0 = 8       (bits 127:112)
      g1[4] = (int)(16u | (32u << 16));   // tile_dim1 = 16, tile_dim2 = 32
      g1[5] = 128;                        // tensor_dim0_stride = 128 (bits 207:160)
      g1[6] = (int)(16384u << 16);        // tensor_dim1_stride = 16384 (bits 255:208)
      g1[7] = 0;
      v4i g2;
      g2[0] = 128;                        // tensor_dim2 = 128
      g2[1] = 0; g2[2] = 0; g2[3] = 0;    // no dim3 / iterate
      v4i g3 = {0, 0, 0, 0};
      v8i g4 = {0, 0, 0, 0, 0, 0, 0, 0};  // unused trailing group (clang-23 6-arg form)
      __builtin_amdgcn_tensor_load_to_lds(g0, g1, g2, g3, g4, 0);
      __builtin_amdgcn_s_wait_tensorcnt(0);
    }

    // async completions are per-wave; wait ours, then barrier with all waves
    asm volatile("s_wait_asynccnt 0x0" ::: "memory");
    __syncthreads();

    // ---- WMMA: per wave, 4 blocks x 2 k-steps of V_WMMA_F32_16X16X4_F32
#pragma unroll
    for (int i = 0; i < 4; ++i) {
      const int bl = wave * 4 + i;
      const float* __restrict__ Ab = &lds[bl * ABSTR + laneM * ROWS];              // A row M
      const float* __restrict__ Bb = &lds[32 * ABSTR + bl * ABSTR + laneM * ROWS]; // B col N
#pragma unroll
      for (int kk = 0; kk < 2; ++kk) {
        const int kb = kk * 4 + laneK;  // k = vgpr + 2*(lane>=16)
        v2f a, b;
        a.x = Ab[kb];
        a.y = Ab[kb + 1];
        b.x = Bb[kb];
        b.y = Bb[kb + 1];
        acc[i] = __builtin_amdgcn_wmma_f32_16x16x4_f32(
            false, a, false, b, (short)0, acc[i], false, false);
      }
    }
  }

  // ---- transpose accumulators through LDS so stores are b-contiguous (128B lines)
  __syncthreads();
#pragma unroll
  for (int i = 0; i < 4; ++i) {
    const int bl = wave * 4 + i;
#pragma unroll
    for (int v = 0; v < 8; ++v) {
      const int tt = v + 8 * (lane >> 4);  // C layout: M = vgpr + 8*(lane>=16)
      lds[bl * CBSTR + tt * 17 + laneM] = acc[i][v];
    }
  }
  __syncthreads();
  {
    const int bl = tid & 31, r = tid >> 5;
#pragma unroll
    for (int it = 0; it < 32; ++it) {
      const int p = it * 8 + r;  // 256 (tt,nn) pairs
      const int tt = p >> 4, nn = p & 15;
      float val = lds[bl * CBSTR + tt * 17 + nn];
      if (PHASE == 2) val += bias[(n0 + nn) * SD + (b0 + bl)];
      Out[(size_t)(t0 + tt) * NF + (size_t)(n0 + nn) * SD + (b0 + bl)] = val;
    }
  }
}

extern "C" void kernel_launch(void* const* d_in, const int* in_sizes, int n_in,
                              void* d_out, int out_size, void* d_ws, size_t ws_size,
                              hipStream_t stream) {
  const float* x = (const float*)d_in[0];     // (2, 2048, 16384) f32
  const float* L = (const float*)d_in[1];     // (128, 128, 128) f32
  const float* R = (const float*)d_in[2];     // (128, 128, 128) f32
  const float* bias = (const float*)d_in[3];  // (16384,) f32
  float* out = (float*)d_out;
  float* W1 = (float*)d_ws;  // needs T*16384*4 bytes (268 MB for T=4096)

  const int T = in_sizes[0] / NF;  // 4096 tokens
  dim3 blk(256, 1, 1);
  dim3 grid(T / 16, SD / 16, SD / 32);  // (256, 8, 4)

  monarch_phase<1><<<grid, blk, 0, stream>>>(x, L, nullptr, W1);
  monarch_phase<2><<<grid, blk, 0, stream>>>(W1, R, bias, out);
}